// LS_solver_61529701482764
// MI455X (gfx1250) — compile-verified
//
#include <hip/hip_runtime.h>

typedef __attribute__((ext_vector_type(2))) float v2f;
typedef __attribute__((ext_vector_type(8))) float v8f;

namespace {
constexpr int kM = 56;      // markers
constexpr int kJ = 24;      // joints
constexpr int kRows = 4 * kJ;   // 96 rows of the augmented LHS
constexpr int kPitch = 104;     // LDS row pitch: 2*104 % 64 == 16 -> half-waves hit disjoint banks
}

// One cyclic Jacobi rotation on symmetric A, accumulating into V. p,q are
// compile-time constants at every call site (forceinline) so A/V stay in regs.
__device__ __forceinline__ void jrot(float A[3][3], float V[3][3], int p, int q) {
  float apq = A[p][q];
  if (fabsf(apq) < 1e-25f) return;
  float tau = (A[q][q] - A[p][p]) / (2.0f * apq);
  float t = copysignf(1.0f, tau) / (fabsf(tau) + sqrtf(1.0f + tau * tau));
  float c = 1.0f / sqrtf(1.0f + t * t);
  float s = t * c;
#pragma unroll
  for (int k = 0; k < 3; ++k) {
    float akp = A[k][p], akq = A[k][q];
    A[k][p] = c * akp - s * akq;
    A[k][q] = s * akp + c * akq;
  }
#pragma unroll
  for (int k = 0; k < 3; ++k) {
    float apk = A[p][k], aqk = A[q][k];
    A[p][k] = c * apk - s * aqk;
    A[q][k] = s * apk + c * aqk;
  }
#pragma unroll
  for (int k = 0; k < 3; ++k) {
    float vkp = V[k][p], vkq = V[k][q];
    V[k][p] = c * vkp - s * vkq;
    V[k][q] = s * vkp + c * vkq;
  }
}

__global__ __launch_bounds__(64) void procrustes_wmma_kernel(
    const float* __restrict__ X,   // (B, M, 3)
    const float* __restrict__ Z,   // (B, M, J, 3)
    const int* __restrict__ W,     // (M, J) in {0,1}
    float* __restrict__ out,       // (B, J, 3, 4)
    int Btot) {
  // Per-batch staged operands for the WMMA GEMM O(96x4) = ZA(96x56) * XA(56x4)
  __shared__ float za[2][kM][kPitch];  // za[b][m][4j+a] = w[m,j] * (z0,z1,z2,1)
  __shared__ float xa[2][kM][4];       // xa[b][m][c]    = (x0,x1,x2,1)
  __shared__ float ob[2][kRows][4];    // GEMM result

  const int tid = threadIdx.x;
  const int b0 = blockIdx.x * 2;

  // ---------------- stage Xaug ----------------
  for (int it = tid; it < 2 * kM; it += 64) {
    int bl = it / kM, m = it - bl * kM;
    const float* xp = X + ((size_t)(b0 + bl) * kM + m) * 3;
    xa[bl][m][0] = xp[0];
    xa[bl][m][1] = xp[1];
    xa[bl][m][2] = xp[2];
    xa[bl][m][3] = 1.0f;
  }
  // ---------------- stage masked Zaug (streams Z once, coalesced) ----------------
  for (int it = tid; it < 2 * kM * kJ; it += 64) {
    int bl = it / (kM * kJ);
    int rm = it - bl * (kM * kJ);
    int m = rm / kJ, j = rm - m * kJ;
    float wv = (float)W[m * kJ + j];
    const float* zp = Z + (((size_t)(b0 + bl) * kM + m) * kJ + j) * 3;
    float* dst = &za[bl][m][4 * j];
    dst[0] = wv * zp[0];
    dst[1] = wv * zp[1];
    dst[2] = wv * zp[2];
    dst[3] = wv;
  }
  __syncthreads();

  // ---------------- per-wave WMMA reduction (one wave = one batch) ----------------
  {
    const int wave = tid >> 5;  // batch b0 + wave
    const int lane = tid & 31;
    const int hi = lane >> 4;     // half-wave selector (K offset per ISA A layout)
    const int n = lane & 15;      // column / row-in-tile
    const int nsel = (n < 4) ? n : 0;           // branchless operand select:
    const float nmask = (n < 4) ? 1.0f : 0.0f;  // EXEC stays all-ones for WMMA

    v8f a0 = {}, a1 = {}, a2 = {}, a3 = {}, a4 = {}, a5 = {};
    for (int k0 = 0; k0 < kM; k0 += 4) {
      const int ko = k0 + 2 * hi;
      // B fragment (4x16): rows K=ko, ko+1 for this half-wave, cols 4..15 zero
      v2f bf;
      bf.x = xa[wave][ko][nsel] * nmask;
      bf.y = xa[wave][ko + 1][nsel] * nmask;
      const float* r0 = &za[wave][ko][0];
      const float* r1 = &za[wave][ko + 1][0];
      v2f af;
      // A fragments: row = tile*16 + n, VGPR0/1 hold K=ko / ko+1 (ISA 16x4 f32 layout)
      af.x = r0[n + 0];  af.y = r1[n + 0];
      a0 = __builtin_amdgcn_wmma_f32_16x16x4_f32(false, af, false, bf, (short)0, a0, false, false);
      af.x = r0[n + 16]; af.y = r1[n + 16];
      a1 = __builtin_amdgcn_wmma_f32_16x16x4_f32(false, af, false, bf, (short)0, a1, false, false);
      af.x = r0[n + 32]; af.y = r1[n + 32];
      a2 = __builtin_amdgcn_wmma_f32_16x16x4_f32(false, af, false, bf, (short)0, a2, false, false);
      af.x = r0[n + 48]; af.y = r1[n + 48];
      a3 = __builtin_amdgcn_wmma_f32_16x16x4_f32(false, af, false, bf, (short)0, a3, false, false);
      af.x = r0[n + 64]; af.y = r1[n + 64];
      a4 = __builtin_amdgcn_wmma_f32_16x16x4_f32(false, af, false, bf, (short)0, a4, false, false);
      af.x = r0[n + 80]; af.y = r1[n + 80];
      a5 = __builtin_amdgcn_wmma_f32_16x16x4_f32(false, af, false, bf, (short)0, a5, false, false);
    }
    // D layout (16x16 f32): VGPR i, half-wave hi -> row i + 8*hi, col n. Keep cols 0..3.
    if (n < 4) {
#pragma unroll
      for (int i = 0; i < 8; ++i) {
        int r = i + 8 * hi;
        ob[wave][r + 0][n] = a0[i];
        ob[wave][r + 16][n] = a1[i];
        ob[wave][r + 32][n] = a2[i];
        ob[wave][r + 48][n] = a3[i];
        ob[wave][r + 64][n] = a4[i];
        ob[wave][r + 80][n] = a5[i];
      }
    }
  }
  __syncthreads();

  // ---------------- per-(batch, joint) Kabsch ----------------
  if (tid < 2 * kJ) {
    const int bl = tid / kJ, j = tid - bl * kJ;
    const int b = b0 + bl;

    float zs[3], xs[3];
    float H[3][3];
    const float cnt = ob[bl][4 * j + 3][3];
    const float inv = 1.0f / cnt;
#pragma unroll
    for (int c = 0; c < 3; ++c) xs[c] = ob[bl][4 * j + 3][c];
#pragma unroll
    for (int a = 0; a < 3; ++a) {
      zs[a] = ob[bl][4 * j + a][3];
#pragma unroll
      for (int c = 0; c < 3; ++c)
        H[a][c] = ob[bl][4 * j + a][c] - zs[a] * xs[c] * inv;
    }

    // Symmetric A = H^T H, eigenvectors via cyclic Jacobi -> V (right singular vecs)
    float A[3][3];
#pragma unroll
    for (int i = 0; i < 3; ++i)
#pragma unroll
      for (int k = 0; k < 3; ++k)
        A[i][k] = H[0][i] * H[0][k] + H[1][i] * H[1][k] + H[2][i] * H[2][k];
    float V[3][3] = {{1.f, 0.f, 0.f}, {0.f, 1.f, 0.f}, {0.f, 0.f, 1.f}};
#pragma unroll
    for (int sweep = 0; sweep < 4; ++sweep) {
      jrot(A, V, 0, 1);
      jrot(A, V, 0, 2);
      jrot(A, V, 1, 2);
    }
    // sort eigenpairs descending
    float lam[3] = {A[0][0], A[1][1], A[2][2]};
    auto cswap = [&](int p, int q) {
      if (lam[p] < lam[q]) {
        float tl = lam[p]; lam[p] = lam[q]; lam[q] = tl;
#pragma unroll
        for (int k = 0; k < 3; ++k) {
          float tv = V[k][p]; V[k][p] = V[k][q]; V[k][q] = tv;
        }
      }
    };
    cswap(0, 1); cswap(0, 2); cswap(1, 2);

    // Left singular vectors: u0 = norm(H v0); u1 = GS(H v1); u2 = u0 x u1 (det(U')=+1).
    // R = V diag(1,1,d) U'^T with d = sign(det V) is exactly the reference rotation.
    float u0[3], u1[3], u2[3];
#pragma unroll
    for (int a = 0; a < 3; ++a)
      u0[a] = H[a][0] * V[0][0] + H[a][1] * V[1][0] + H[a][2] * V[2][0];
    float n0 = sqrtf(u0[0] * u0[0] + u0[1] * u0[1] + u0[2] * u0[2]);
    float i0 = 1.0f / fmaxf(n0, 1e-30f);
#pragma unroll
    for (int a = 0; a < 3; ++a) u0[a] *= i0;
#pragma unroll
    for (int a = 0; a < 3; ++a)
      u1[a] = H[a][0] * V[0][1] + H[a][1] * V[1][1] + H[a][2] * V[2][1];
    float pr = u0[0] * u1[0] + u0[1] * u1[1] + u0[2] * u1[2];
#pragma unroll
    for (int a = 0; a < 3; ++a) u1[a] -= pr * u0[a];
    float n1 = sqrtf(u1[0] * u1[0] + u1[1] * u1[1] + u1[2] * u1[2]);
    float i1 = 1.0f / fmaxf(n1, 1e-30f);
#pragma unroll
    for (int a = 0; a < 3; ++a) u1[a] *= i1;
    u2[0] = u0[1] * u1[2] - u0[2] * u1[1];
    u2[1] = u0[2] * u1[0] - u0[0] * u1[2];
    u2[2] = u0[0] * u1[1] - u0[1] * u1[0];

    float detV = V[0][0] * (V[1][1] * V[2][2] - V[1][2] * V[2][1]) -
                 V[0][1] * (V[1][0] * V[2][2] - V[1][2] * V[2][0]) +
                 V[0][2] * (V[1][0] * V[2][1] - V[1][1] * V[2][0]);
    float d = (detV < 0.0f) ? -1.0f : 1.0f;

    float R[3][3];
#pragma unroll
    for (int c = 0; c < 3; ++c)
#pragma unroll
      for (int a = 0; a < 3; ++a)
        R[c][a] = V[c][0] * u0[a] + V[c][1] * u1[a] + d * V[c][2] * u2[a];

    float* op = out + ((size_t)b * kJ + j) * 12;
#pragma unroll
    for (int c = 0; c < 3; ++c) {
      float zm0 = zs[0] * inv, zm1 = zs[1] * inv, zm2 = zs[2] * inv;
      float tc = xs[c] * inv - (R[c][0] * zm0 + R[c][1] * zm1 + R[c][2] * zm2);
      op[c * 4 + 0] = R[c][0];
      op[c * 4 + 1] = R[c][1];
      op[c * 4 + 2] = R[c][2];
      op[c * 4 + 3] = tc;
    }
  }
}

extern "C" void kernel_launch(void* const* d_in, const int* in_sizes, int n_in,
                              void* d_out, int out_size, void* d_ws, size_t ws_size,
                              hipStream_t stream) {
  const float* X = (const float*)d_in[0];  // (B, 56, 3) f32
  const float* Z = (const float*)d_in[1];  // (B, 56, 24, 3) f32
  const int* W = (const int*)d_in[2];      // (56, 24) i32
  float* out = (float*)d_out;              // (B, 24, 3, 4) f32

  const int Btot = in_sizes[0] / (kM * 3);  // 8192
  const int blocks = Btot / 2;              // 2 batches (one per wave32) per block
  procrustes_wmma_kernel<<<blocks, 64, 0, stream>>>(X, Z, W, out, Btot);
}